// CollisionNet_9457517986452
// MI455X (gfx1250) — compile-verified
//
#include <hip/hip_runtime.h>

typedef float v2f __attribute__((ext_vector_type(2)));
typedef float v8f __attribute__((ext_vector_type(8)));
typedef int   v4i __attribute__((vector_size(16)));

#define PAIRS_PER_TILE 64
#define ROWS_PER_TILE  128
#define BLOCK_THREADS  128
#define XS  100   // X/H row stride (floats): conflict-free WMMA A-loads (b64)
#define WSF 224   // interleaved weight K-pair row stride in floats (112 float2)

// float-index offsets into dynamic LDS
#define OFF_W1 0                               // 48*224 = 10752
#define OFF_W2 (48*WSF)                        // +10752
#define OFF_X  (2*48*WSF)                      // 21504 ; 128*100 = 12800
#define OFF_T  (OFF_X + ROWS_PER_TILE*XS)      // 34304 ; 64*16 = 1024
#define OFF_TI (OFF_T + PAIRS_PER_TILE*16)     // 35328 ; 64*12 = 768
#define OFF_B1 (OFF_TI + PAIRS_PER_TILE*12)    // 36096
#define OFF_B2 (OFF_B1 + 96)
#define OFF_W3 (OFF_B2 + 96)
#define OFF_Y  (OFF_W3 + 96)
#define OFF_C  (OFF_Y + ROWS_PER_TILE)
#define LDS_FLOATS (OFF_C + 4)                 // 36516 floats = 146,064 B

#if __has_builtin(__builtin_amdgcn_global_load_async_to_lds_b128)
#define HAVE_ASYNC_LDS 1
#else
#define HAVE_ASYNC_LDS 0
#endif

typedef v4i __attribute__((address_space(1)))* g1p4;
typedef v4i __attribute__((address_space(3)))* l3p4;
// Global generic address == global address; LDS generic addr[31:0] == LDS offset (ISA 10.2).
#define TO_G1(p) ((g1p4)(unsigned long long)(const void*)(p))
#define TO_L3(p) ((l3p4)(unsigned int)(unsigned long long)(const void*)(p))

// Row-major 4x4 inverse via adjugate; emits only rows 0..2 (12 floats, stride 4).
__device__ __forceinline__ void invert4x4_rows012(const float* m, float* o) {
    float i0  =  m[5]*m[10]*m[15] - m[5]*m[11]*m[14] - m[9]*m[6]*m[15] + m[9]*m[7]*m[14] + m[13]*m[6]*m[11] - m[13]*m[7]*m[10];
    float i4  = -m[4]*m[10]*m[15] + m[4]*m[11]*m[14] + m[8]*m[6]*m[15] - m[8]*m[7]*m[14] - m[12]*m[6]*m[11] + m[12]*m[7]*m[10];
    float i8  =  m[4]*m[9]*m[15]  - m[4]*m[11]*m[13] - m[8]*m[5]*m[15] + m[8]*m[7]*m[13] + m[12]*m[5]*m[11] - m[12]*m[7]*m[9];
    float i12 = -m[4]*m[9]*m[14]  + m[4]*m[10]*m[13] + m[8]*m[5]*m[14] - m[8]*m[6]*m[13] - m[12]*m[5]*m[10] + m[12]*m[6]*m[9];
    float i1  = -m[1]*m[10]*m[15] + m[1]*m[11]*m[14] + m[9]*m[2]*m[15] - m[9]*m[3]*m[14] - m[13]*m[2]*m[11] + m[13]*m[3]*m[10];
    float i5  =  m[0]*m[10]*m[15] - m[0]*m[11]*m[14] - m[8]*m[2]*m[15] + m[8]*m[3]*m[14] + m[12]*m[2]*m[11] - m[12]*m[3]*m[10];
    float i9  = -m[0]*m[9]*m[15]  + m[0]*m[11]*m[13] + m[8]*m[1]*m[15] - m[8]*m[3]*m[13] - m[12]*m[1]*m[11] + m[12]*m[3]*m[9];
    float i2  =  m[1]*m[6]*m[15]  - m[1]*m[7]*m[14]  - m[5]*m[2]*m[15] + m[5]*m[3]*m[14] + m[13]*m[2]*m[7]  - m[13]*m[3]*m[6];
    float i6  = -m[0]*m[6]*m[15]  + m[0]*m[7]*m[14]  + m[4]*m[2]*m[15] - m[4]*m[3]*m[14] - m[12]*m[2]*m[7]  + m[12]*m[3]*m[6];
    float i10 =  m[0]*m[5]*m[15]  - m[0]*m[7]*m[13]  - m[4]*m[1]*m[15] + m[4]*m[3]*m[13] + m[12]*m[1]*m[7]  - m[12]*m[3]*m[5];
    float i3  = -m[1]*m[6]*m[11]  + m[1]*m[7]*m[10]  + m[5]*m[2]*m[11] - m[5]*m[3]*m[10] - m[9]*m[2]*m[7]   + m[9]*m[3]*m[6];
    float i7  =  m[0]*m[6]*m[11]  - m[0]*m[7]*m[10]  - m[4]*m[2]*m[11] + m[4]*m[3]*m[10] + m[8]*m[2]*m[7]   - m[8]*m[3]*m[6];
    float i11 = -m[0]*m[5]*m[11]  + m[0]*m[7]*m[9]   + m[4]*m[1]*m[11] - m[4]*m[3]*m[9]  - m[8]*m[1]*m[7]   + m[8]*m[3]*m[5];
    float det = m[0]*i0 + m[1]*i4 + m[2]*i8 + m[3]*i12;
    float r = 1.0f / det;
    o[0]=i0*r;  o[1]=i1*r;  o[2]=i2*r;  o[3]=i3*r;
    o[4]=i4*r;  o[5]=i5*r;  o[6]=i6*r;  o[7]=i7*r;
    o[8]=i8*r;  o[9]=i9*r;  o[10]=i10*r; o[11]=i11*r;
}

__global__ void __launch_bounds__(BLOCK_THREADS)
collision_mlp_wmma(const int* __restrict__ o1, const int* __restrict__ o2,
                   const float* __restrict__ T, const float* __restrict__ geoms,
                   const float* __restrict__ W1, const float* __restrict__ b1,
                   const float* __restrict__ W2, const float* __restrict__ b2,
                   const float* __restrict__ W3, const float* __restrict__ b3,
                   float* __restrict__ out, int npairs)
{
    extern __shared__ float smem[];
    const int tid = threadIdx.x;

    // ---- Stage weights into LDS once per block (K-pair interleaved: one b64/fragment) ----
    for (int i = tid; i < 48 * 96; i += BLOCK_THREADS) {
        int kp = i / 96, n = i - kp * 96;
        int s0 = (2 * kp) * 96 + n, s1 = s0 + 96;
        int d = kp * WSF + 2 * n;
        smem[OFF_W1 + d] = W1[s0];  smem[OFF_W1 + d + 1] = W1[s1];
        smem[OFF_W2 + d] = W2[s0];  smem[OFF_W2 + d + 1] = W2[s1];
    }
    if (tid < 96) {
        smem[OFF_B1 + tid] = b1[tid];
        smem[OFF_B2 + tid] = b2[tid];
        smem[OFF_W3 + tid] = W3[tid];
    }
    if (tid == 0) smem[OFF_C] = b3[0];
    __syncthreads();

    const int lane = tid & 31;
    const int wave = tid >> 5;       // 0..3: owns rows [16w,16w+16) and [64+16w, 64+16w+16)
    const int half = lane >> 4;      // lane group
    const int lcol = lane & 15;
    const int koff = half << 1;

    const int ntiles = (npairs + PAIRS_PER_TILE - 1) / PAIRS_PER_TILE;

    for (int tile = blockIdx.x; tile < ntiles; tile += gridDim.x) {
        const int p0 = tile * PAIRS_PER_TILE;

        // ---- Stage T (64 pairs = 256 float4) : async global->LDS when available ----
        for (int i = tid; i < PAIRS_PER_TILE * 4; i += BLOCK_THREADS) {
            long g4 = (long)p0 * 4 + i;
            if (g4 < (long)npairs * 4) {
#if HAVE_ASYNC_LDS
                __builtin_amdgcn_global_load_async_to_lds_b128(
                    TO_G1(T + g4 * 4), TO_L3(&smem[OFF_T + i * 4]), 0, 0);
#else
                float4 v = ((const float4*)T)[g4];
                smem[OFF_T + i*4 + 0] = v.x; smem[OFF_T + i*4 + 1] = v.y;
                smem[OFF_T + i*4 + 2] = v.z; smem[OFF_T + i*4 + 3] = v.w;
#endif
            }
        }
#if HAVE_ASYNC_LDS
#if __has_builtin(__builtin_amdgcn_s_wait_asynccnt)
        __builtin_amdgcn_s_wait_asynccnt(0);
#else
        asm volatile("s_wait_asynccnt 0x0" ::: "memory");
#endif
#endif
        __syncthreads();

        // ---- Per-pair 4x4 inverse (rows 0..2) ----
        if (tid < PAIRS_PER_TILE && (p0 + tid) < npairs)
            invert4x4_rows012(&smem[OFF_T + tid * 16], &smem[OFF_TI + tid * 12]);
        __syncthreads();

        // ---- Build 128 input rows: [geoms[a] | transform(geoms[b], M)] (1 thread/row) ----
        {
            int r = tid, p = r >> 1, br = r & 1, gp = p0 + p;
            if (gp < npairs) {
                int ia = br ? o2[gp] : o1[gp];
                int ib = br ? o1[gp] : o2[gp];
                float* Xr = &smem[OFF_X + r * XS];

                const float4* ga = (const float4*)(geoms + (size_t)ia * 48);
                #pragma unroll
                for (int q = 0; q < 12; ++q) {
                    float4 v = ga[q];
                    Xr[q*4+0] = v.x; Xr[q*4+1] = v.y; Xr[q*4+2] = v.z; Xr[q*4+3] = v.w;
                }

                float gb[48];
                const float4* gB = (const float4*)(geoms + (size_t)ib * 48);
                #pragma unroll
                for (int q = 0; q < 12; ++q) {
                    float4 v = gB[q];
                    gb[q*4+0] = v.x; gb[q*4+1] = v.y; gb[q*4+2] = v.z; gb[q*4+3] = v.w;
                }

                const float* R = br ? &smem[OFF_TI + p * 12] : &smem[OFF_T + p * 16];
                float r00=R[0], r01=R[1], r02=R[2],  t0=R[3];
                float r10=R[4], r11=R[5], r12=R[6],  t1=R[7];
                float r20=R[8], r21=R[9], r22=R[10], t2=R[11];
                #pragma unroll
                for (int mv = 0; mv < 16; ++mv) {
                    float v0 = gb[3*mv], v1 = gb[3*mv+1], v2 = gb[3*mv+2];
                    Xr[48 + 3*mv + 0] = fmaf(r00, v0, fmaf(r01, v1, fmaf(r02, v2, t0)));
                    Xr[48 + 3*mv + 1] = fmaf(r10, v0, fmaf(r11, v1, fmaf(r12, v2, t1)));
                    Xr[48 + 3*mv + 2] = fmaf(r20, v0, fmaf(r21, v1, fmaf(r22, v2, t2)));
                }
            }
        }
        __syncthreads();

        // ---- Two 96x96 layers on the matrix pipe; 2 M-tiles per wave ----
        const float* A0 = &smem[OFF_X + (wave * 16 + lcol) * XS];
        const float* A1 = A0 + 64 * XS;

        #pragma unroll
        for (int layer = 0; layer < 2; ++layer) {
            const int wOff = layer ? OFF_W2 : OFF_W1;
            const int bOff = layer ? OFF_B2 : OFF_B1;

            v8f acc0[6], acc1[6];
            #pragma unroll
            for (int n = 0; n < 6; ++n) {
                float bv = smem[bOff + n * 16 + lcol];
                #pragma unroll
                for (int j = 0; j < 8; ++j) { acc0[n][j] = bv; acc1[n][j] = bv; }
            }

            #pragma unroll 2
            for (int k4 = 0; k4 < 96; k4 += 4) {
                v2f a0 = *(const v2f*)(A0 + k4 + koff);
                v2f a1 = *(const v2f*)(A1 + k4 + koff);
                const float* wb = &smem[wOff + ((k4 >> 1) + half) * WSF + (lcol << 1)];
                #pragma unroll
                for (int n = 0; n < 6; ++n) {
                    v2f bb = *(const v2f*)(wb + (n << 5));
                    acc0[n] = __builtin_amdgcn_wmma_f32_16x16x4_f32(
                        false, a0, false, bb, (short)0, acc0[n], false, false);
                    acc1[n] = __builtin_amdgcn_wmma_f32_16x16x4_f32(
                        false, a1, false, bb, (short)0, acc1[n], false, false);
                }
            }

            // ReLU + write H back into the X buffer (wave-private rows; no barrier)
            #pragma unroll
            for (int n = 0; n < 6; ++n) {
                #pragma unroll
                for (int j = 0; j < 8; ++j) {
                    int row0 = wave * 16 + j + 8 * half;
                    smem[OFF_X + row0 * XS + n * 16 + lcol]        = fmaxf(acc0[n][j], 0.0f);
                    smem[OFF_X + (row0 + 64) * XS + n * 16 + lcol] = fmaxf(acc1[n][j], 0.0f);
                }
            }
        }
        __syncthreads();

        // ---- Final 96->1 layer (1 thread/row), then average branches ----
        {
            const float* h = &smem[OFF_X + tid * XS];
            float acc = smem[OFF_C];
            #pragma unroll 8
            for (int c = 0; c < 96; ++c) acc = fmaf(h[c], smem[OFF_W3 + c], acc);
            smem[OFF_Y + tid] = acc;
        }
        __syncthreads();
        if (tid < PAIRS_PER_TILE) {
            int gp = p0 + tid;
            if (gp < npairs)
                out[gp] = 0.5f * (smem[OFF_Y + 2 * tid] + smem[OFF_Y + 2 * tid + 1]);
        }
        __syncthreads();
    }
}

extern "C" void kernel_launch(void* const* d_in, const int* in_sizes, int n_in,
                              void* d_out, int out_size, void* d_ws, size_t ws_size,
                              hipStream_t stream) {
    const int*   o1    = (const int*)d_in[0];
    const int*   o2    = (const int*)d_in[1];
    const float* T     = (const float*)d_in[2];
    const float* geoms = (const float*)d_in[3];
    const float* W1    = (const float*)d_in[4];
    const float* b1    = (const float*)d_in[5];
    const float* W2    = (const float*)d_in[6];
    const float* b2    = (const float*)d_in[7];
    const float* W3    = (const float*)d_in[8];
    const float* b3    = (const float*)d_in[9];
    float* out = (float*)d_out;

    int npairs = in_sizes[0];
    int ntiles = (npairs + PAIRS_PER_TILE - 1) / PAIRS_PER_TILE;
    int blocks = ntiles < 4096 ? ntiles : 4096;
    size_t lds_bytes = (size_t)LDS_FLOATS * sizeof(float);

    (void)hipFuncSetAttribute((const void*)collision_mlp_wmma,
                              hipFuncAttributeMaxDynamicSharedMemorySize,
                              (int)lds_bytes);

    collision_mlp_wmma<<<blocks, BLOCK_THREADS, lds_bytes, stream>>>(
        o1, o2, T, geoms, W1, b1, W2, b2, W3, b3, out, npairs);
}